// GCN_31585189495371
// MI455X (gfx1250) — compile-verified
//
#include <hip/hip_runtime.h>

typedef float v2f __attribute__((ext_vector_type(2)));
typedef float v8f __attribute__((ext_vector_type(8)));

#define G_    256
#define NPG_  400
#define DEG_  32
#define EPG_  (NPG_ * DEG_)     // 12800 edges per graph
#define FIN_  400
#define H_    64
#define N_    (G_ * NPG_)       // 102400
#define E_    (N_ * DEG_)       // 3276800

// ---------------------------------------------------------------------------
// deg[n] = sum(ew over edges with dst==n) + 1 (self loop); dinv = rsqrt(deg)
// one block per graph, degree accumulated in LDS (edges never cross graphs)
// ---------------------------------------------------------------------------
__global__ void k_dinv(const int* __restrict__ ei, const float* __restrict__ ew,
                       float* __restrict__ dinv) {
    __shared__ float sdeg[NPG_];
    const int g = blockIdx.x, t = threadIdx.x;
    const int base = g * NPG_;
    const int e0 = g * EPG_;
    for (int i = t; i < NPG_; i += blockDim.x) sdeg[i] = 0.0f;
    __syncthreads();
    for (int e = t; e < EPG_; e += blockDim.x) {
        int dst = ei[E_ + e0 + e];
        atomicAdd(&sdeg[dst - base], ew[e0 + e]);
    }
    __syncthreads();
    for (int i = t; i < NPG_; i += blockDim.x)
        dinv[base + i] = rsqrtf(sdeg[i] + 1.0f);   // deg >= 1 always
}

// ---------------------------------------------------------------------------
// norm[e] = dinv[src] * w[e] * dinv[dst]   (precomputed once, reused 3 layers)
// ---------------------------------------------------------------------------
__global__ void k_norm(const int* __restrict__ ei, const float* __restrict__ ew,
                       const float* __restrict__ dinv, float* __restrict__ nrm) {
    int e = blockIdx.x * blockDim.x + threadIdx.x;
    if (e < E_) {
        int s = ei[e], d = ei[E_ + e];
        nrm[e] = dinv[s] * ew[e] * dinv[d];
    }
}

// ---------------------------------------------------------------------------
// fp32 WMMA GEMM:  C[M x 64] = A[M x K] (row major, lda=K) @ B[K x 64]
// Block = 128 threads (4 waves), block tile = 64 rows x 64 cols.
// Wave w owns rows [16w,16w+16) and ALL 64 cols via 4 accumulators, so each
// A fragment (1 ds_load_b64) feeds 4 V_WMMA_F32_16X16X4_F32.
// A tile:  LDS, rows padded to 162 dwords -> conflict-free b64 reads.
// B tile:  LDS, pre-swizzled into fragment-pair order keyed by
//          (k-step, col-tile, lane) -> wave reads are lane-stride-2 dwords
//          across all 64 banks, conflict-free, one ds_load_b64 per fragment.
// ---------------------------------------------------------------------------
#define KC   160                 // K chunk (multiple of 16)
#define APAD 162                 // padded A row stride in dwords

__global__ void k_gemm(const float* __restrict__ A, const float* __restrict__ B,
                       float* __restrict__ C, const float* __restrict__ bias,
                       int K) {
    __shared__ float As[64 * APAD];          // 41472 B
    __shared__ float Bs[KC * 64];            // 40960 B (pairs: [k4][c][lane][2])
    const int row0 = blockIdx.x * 64;
    const int tid  = threadIdx.x;
    const int lane = tid & 31;
    const int wv   = tid >> 5;               // wave -> rows [16wv, 16wv+16)
    const int mn   = lane & 15;
    const int kh   = lane >> 4;              // K-half selector (16x4 f32 layout)

    // fixed (c,l) slot this thread stages for B each k-step
    const int sc = (tid >> 5) & 3;           // col tile 0..3
    const int sl = tid & 31;                 // lane slot
    const int skh = sl >> 4, smn = sl & 15;
    const int sn = sc * 16 + smn;

    v8f acc0 = {}, acc1 = {}, acc2 = {}, acc3 = {};

    for (int kb = 0; kb < K; kb += KC) {
        const int kc = (K - kb) < KC ? (K - kb) : KC;   // multiple of 16
        // stage A tile (64 x kc), coalesced rows
        for (int i = tid; i < 64 * kc; i += 128) {
            int r = i / kc, c = i - r * kc;
            As[r * APAD + c] = A[(long)(row0 + r) * K + kb + c];
        }
        // stage B fragments: pair p=(k4,c,l) holds B[k+2kh][n], B[k+2kh+1][n]
        const int nk4 = kc >> 2;
        for (int k4 = 0; k4 < nk4; ++k4) {
            int krow = kb + 4 * k4 + 2 * skh;
            int p = (k4 * 4 + sc) * 32 + sl;
            Bs[p * 2 + 0] = B[(long)krow * 64 + sn];
            Bs[p * 2 + 1] = B[(long)(krow + 1) * 64 + sn];
        }
        __syncthreads();
        const float* ap = &As[(wv * 16 + mn) * APAD + 2 * kh];
        #pragma unroll 2
        for (int k4 = 0; k4 < nk4; ++k4) {
            v2f a = *(const v2f*)(ap + 4 * k4);
            const float* bp = &Bs[(k4 * 4) * 64 + lane * 2];
            v2f b0 = *(const v2f*)(bp + 0);
            v2f b1 = *(const v2f*)(bp + 64);
            v2f b2 = *(const v2f*)(bp + 128);
            v2f b3 = *(const v2f*)(bp + 192);
            acc0 = __builtin_amdgcn_wmma_f32_16x16x4_f32(false, a, false, b0, (short)0, acc0, false, false);
            acc1 = __builtin_amdgcn_wmma_f32_16x16x4_f32(false, a, false, b1, (short)0, acc1, false, false);
            acc2 = __builtin_amdgcn_wmma_f32_16x16x4_f32(false, a, false, b2, (short)0, acc2, false, false);
            acc3 = __builtin_amdgcn_wmma_f32_16x16x4_f32(false, a, false, b3, (short)0, acc3, false, false);
        }
        __syncthreads();
    }
    // D layout: vgpr r -> m = r + 8*kh, n = 16*c + mn
    const int rbase = row0 + wv * 16 + 8 * kh;
    #pragma unroll
    for (int c = 0; c < 4; ++c) {
        const v8f* accp = (c == 0) ? &acc0 : (c == 1) ? &acc1 : (c == 2) ? &acc2 : &acc3;
        int n = c * 16 + mn;
        float bv = bias ? bias[n] : 0.0f;
        #pragma unroll
        for (int r = 0; r < 8; ++r)
            C[(long)(rbase + r) * 64 + n] = (*accp)[r] + bv;
    }
}

// ---------------------------------------------------------------------------
// GCN aggregation for one (graph, 16-feature chunk):
//   out[d] = sum_edges norm * h[s]  +  dinv[i]^2 * h[i]  + bias ; optional relu
// sh_in + sh_out = 50 KB LDS; scatter via LDS float atomics (ds_add_f32).
// grid = (256 graphs, 4 feature chunks), block = 256 (fx = tid&15 feature,
// ey = tid>>4 -> 16 edges in flight)
// ---------------------------------------------------------------------------
__global__ void k_agg(const float* __restrict__ hin, const int* __restrict__ ei,
                      const float* __restrict__ nrm, const float* __restrict__ dinv,
                      const float* __restrict__ bias, float* __restrict__ hout,
                      int do_relu) {
    __shared__ float sin_[NPG_ * 16];
    __shared__ float sout[NPG_ * 16];
    const int g = blockIdx.x, fc = blockIdx.y;
    const int base = g * NPG_;
    const int fbase = fc * 16;
    const int tid = threadIdx.x;
    const int fx = tid & 15, ey = tid >> 4;

    for (int i = tid; i < NPG_ * 16; i += 256) {
        int node = i >> 4, f = i & 15;
        sin_[i] = hin[(long)(base + node) * 64 + fbase + f];
        sout[i] = 0.0f;
    }
    __syncthreads();

    const int e0 = g * EPG_;
    for (int e = ey; e < EPG_; e += 16) {
        int ge = e0 + e;
        int s = ei[ge] - base;
        int d = ei[E_ + ge] - base;
        float w = nrm[ge];
        atomicAdd(&sout[d * 16 + fx], w * sin_[s * 16 + fx]);
    }
    __syncthreads();

    for (int i = tid; i < NPG_ * 16; i += 256) {
        int node = i >> 4, f = i & 15;
        float di = dinv[base + node];
        float v = sout[i] + di * di * sin_[i] + bias[fbase + f];
        if (do_relu) v = fmaxf(v, 0.0f);
        hout[(long)(base + node) * 64 + fbase + f] = v;
    }
}

// ---------------------------------------------------------------------------
// out[g, 0:2] = gbuf[g, 0:64] @ Wl[64 x 2] + bl   (tiny head)
// ---------------------------------------------------------------------------
__global__ void k_head(const float* __restrict__ gbuf, const float* __restrict__ Wl,
                       const float* __restrict__ bl, float* __restrict__ out) {
    int g = blockIdx.x * blockDim.x + threadIdx.x;
    if (g < G_) {
        float a0 = bl[0], a1 = bl[1];
        #pragma unroll 8
        for (int f = 0; f < 64; ++f) {
            float v = gbuf[g * 64 + f];
            a0 += v * Wl[f * 2 + 0];
            a1 += v * Wl[f * 2 + 1];
        }
        out[g * 2 + 0] = a0;
        out[g * 2 + 1] = a1;
    }
}

// ---------------------------------------------------------------------------
extern "C" void kernel_launch(void* const* d_in, const int* in_sizes, int n_in,
                              void* d_out, int out_size, void* d_ws, size_t ws_size,
                              hipStream_t stream) {
    const float* x  = (const float*)d_in[0];
    const int*   ei = (const int*)  d_in[1];
    const float* ew = (const float*)d_in[2];
    const float* W1 = (const float*)d_in[3];  const float* b1 = (const float*)d_in[4];
    const float* W2 = (const float*)d_in[5];  const float* b2 = (const float*)d_in[6];
    const float* W3 = (const float*)d_in[7];  const float* b3 = (const float*)d_in[8];
    const float* Wc = (const float*)d_in[9];  const float* bc = (const float*)d_in[10];
    const float* Wl = (const float*)d_in[11]; const float* bl = (const float*)d_in[12];

    char* ws = (char*)d_ws;
    float* dinv = (float*)(ws + 0);                                // N_ floats
    float* nrm  = (float*)(ws + 409600);                           // E_ floats
    float* bufA = (float*)(ws + 409600 + 13107200);                // N_*64
    float* bufB = (float*)(ws + 409600 + 13107200 + 26214400);     // N_*64
    float* gbuf = (float*)(ws + 409600 + 13107200 + 2 * 26214400); // 256*64

    // normalization
    k_dinv<<<G_, 256, 0, stream>>>(ei, ew, dinv);
    k_norm<<<(E_ + 255) / 256, 256, 0, stream>>>(ei, ew, dinv, nrm);

    dim3 agrid(G_, 4);

    // layer 1: h = relu(agg(x @ W1) + b1)
    k_gemm<<<N_ / 64, 128, 0, stream>>>(x, W1, bufA, nullptr, FIN_);
    k_agg <<<agrid, 256, 0, stream>>>(bufA, ei, nrm, dinv, b1, bufB, 1);

    // layer 2
    k_gemm<<<N_ / 64, 128, 0, stream>>>(bufB, W2, bufA, nullptr, H_);
    k_agg <<<agrid, 256, 0, stream>>>(bufA, ei, nrm, dinv, b2, bufB, 1);

    // layer 3 (no relu)
    k_gemm<<<N_ / 64, 128, 0, stream>>>(bufB, W3, bufA, nullptr, H_);
    k_agg <<<agrid, 256, 0, stream>>>(bufA, ei, nrm, dinv, b3, bufB, 0);

    // head: g = h.reshape(256, 25600) @ Wc + bc ;  out = g @ Wl + bl
    k_gemm<<<G_ / 64, 128, 0, stream>>>(bufB, Wc, gbuf, bc, NPG_ * H_);
    k_head<<<1, 256, 0, stream>>>(gbuf, Wl, bl, (float*)d_out);
}